// FeedbackDelayNetwork_77326591197207
// MI455X (gfx1250) — compile-verified
//
#include <hip/hip_runtime.h>

// ---------------------------------------------------------------------------
// FDN reverb for MI455X (gfx1250, wave32).
//   K1: per-bin 8x8 complex solve (one bin per lane, matrix in LDS)
//   K2: direct real iDFT (spectrum chunked through LDS, exact integer phase)
//   K3: zero-padded input image
//   K4: conv-as-band-Toeplitz GEMM on V_WMMA_F32_16X16X4_F32; padded IR
//       brought into LDS by the Tensor Data Mover (tensor_load_to_lds),
//       B operand fetched as global_load_b128.
// ---------------------------------------------------------------------------

typedef __attribute__((ext_vector_type(2))) float v2f;
typedef __attribute__((ext_vector_type(4))) float f32x4;
typedef __attribute__((ext_vector_type(8))) float v8f;
typedef __attribute__((ext_vector_type(4))) unsigned int u32x4;
typedef __attribute__((ext_vector_type(4))) int i32x4;
typedef __attribute__((ext_vector_type(8))) int i32x8;

struct c32 { float re, im; };
__device__ __forceinline__ c32 cmul(c32 a, c32 b) {
  return { a.re * b.re - a.im * b.im, a.re * b.im + a.im * b.re };
}
__device__ __forceinline__ c32 csub(c32 a, c32 b) { return { a.re - b.re, a.im - b.im }; }
__device__ __forceinline__ c32 cdiv(c32 a, c32 b) {
  float id = 1.0f / (b.re * b.re + b.im * b.im);
  return { (a.re * b.re + a.im * b.im) * id, (a.im * b.re - a.re * b.im) * id };
}

#define SRATE       16000
#define NFFT        32000
#define NBINS       16001   // NFFT/2 + 1
#define NDLY        8
#define NAP         4
#define EARLY_LEN   200
#define TWO_PI_F    6.28318530717958647692f
#define LN10_F      2.30258509299404568402f

// Workspace layout (float units):
//   [0      , 32002)  transfer (float2[16001])
//   [32768  , 64832)  hpad: 16 zeros | 32000 IR | 48 zeros  (32064 floats)
//   [65536  , ... )   xpad: 4 batches x 65536; x[b][i] at xpad[b*65536 + 16000 + i]
#define WS_HPAD_OFF 32768
#define WS_XPAD_OFF 65536
#define XPAD_STRIDE 65536
#define XPAD_LEFT   16000
#define HPAD_LEN    32064
#define CONV_LEFT   15999   // (K-1)//2, K = 32000
#define NQ          8004    // tap-sweep iterations (c = 4q)

// ---------------------------------------------------------------------------
// K1: frequency response. One bin per thread; 8x8 complex system in LDS.
// ---------------------------------------------------------------------------
__global__ void fdn_freq_kernel(const float* __restrict__ in_g,
                                const float* __restrict__ out_g,
                                const float* __restrict__ t0p,
                                const float* __restrict__ alphap,
                                const float* __restrict__ dval,
                                const float* __restrict__ dapl,
                                const float* __restrict__ gapl,
                                float2* __restrict__ transfer) {
  __shared__ c32 Ash[NDLY * NDLY * 64];   // [elem][thread], 32 KB
  const int tid = threadIdx.x;
  const int f = blockIdx.x * 64 + tid;
  if (f >= NBINS) return;
#define AS(r, c) Ash[(((r) * NDLY + (c)) * 64) + tid]

  const float t0 = t0p[0];
  const float alpha = alphap[0];
  const float wk = TWO_PI_F * (float)f / (float)NFFT;
  float swk, cwk;
  __sincosf(wk, &swk, &cwk);
  const c32 zk = { cwk, -swk };           // e^{-j wk}

  c32 dvec[NDLY], filt[NDLY], aptf[NDLY];
#pragma unroll
  for (int d = 0; d < NDLY; ++d) {
    const float dv = dval[d];
    const float fl = floorf(dv);
    // exact phase of e^{-j wk*floor(dv)} via integer residue
    const long long fli = (long long)fl;
    const int m = (int)(((long long)f * fli) % (long long)NFFT);
    float s1, c1;
    __sincosf((float)m * (TWO_PI_F / (float)NFFT), &s1, &c1);
    const c32 zd = { c1, -s1 };
    // fractional-delay allpass interpolator
    const float de = dv - fl;
    const float eta = (1.0f - de) / (1.0f + de);
    const c32 nap = { eta + zk.re, zk.im };
    const c32 dap = { 1.0f + eta * zk.re, eta * zk.im };
    dvec[d] = cmul(zd, cdiv(nap, dap));
    // one-pole absorption filter
    float dsum = dv;
#pragma unroll
    for (int a = 0; a < NAP; ++a) dsum += dapl[d * NAP + a];
    const float dsec = dsum / (float)SRATE;
    const float kk  = __expf(-3.0f * dsec / t0 * LN10_F);
    const float kpi = __expf(-3.0f * dsec / (alpha * t0) * LN10_F);
    const float g = 2.0f * kk * kpi / (kk + kpi);
    const float p = (kk - kpi) / (kk + kpi);
    const c32 fden = { 1.0f - p * zk.re + 1e-8f, -p * zk.im };
    const float id = 1.0f / (fden.re * fden.re + fden.im * fden.im);
    filt[d] = { g * fden.re * id, -g * fden.im * id };
    // allpass decorrelator product; z_da = e^{+j wk*delay} (exact phase via fp64 fmod)
    c32 prod = { 1.0f, 0.0f };
#pragma unroll
    for (int a = 0; a < NAP; ++a) {
      const float dl = dapl[d * NAP + a];
      const double md = fmod((double)f * (double)dl, (double)NFFT);
      float sa, ca;
      __sincosf((float)(md * (double)(TWO_PI_F / (float)NFFT)), &sa, &ca);
      const float ga = gapl[d * NAP + a];
      const c32 nu = { 1.0f + ga * ca, ga * sa };
      const c32 dn = { ga + ca, sa };
      prod = cmul(prod, cdiv(nu, dn));
    }
    aptf[d] = prod;
  }

  // A = I - diag(filt) M diag(aptf) diag(dvec),  M = -I + 0.5
#pragma unroll
  for (int r = 0; r < NDLY; ++r) {
#pragma unroll
    for (int c = 0; c < NDLY; ++c) {
      const float Mrc = (r == c) ? -0.5f : 0.5f;
      c32 fb = cmul(filt[r], aptf[c]);
      fb.re *= Mrc; fb.im *= Mrc;
      const c32 t = cmul(fb, dvec[c]);
      c32 a = { ((r == c) ? 1.0f : 0.0f) - t.re, -t.im };
      AS(r, c) = a;
    }
  }

  c32 bb[NDLY];
#pragma unroll
  for (int r = 0; r < NDLY; ++r) bb[r] = { in_g[r], 0.0f };

  // Gaussian elimination (no pivot), fully unrolled so bb/x stay in VGPRs
#pragma unroll
  for (int p = 0; p < NDLY; ++p) {
    const c32 piv = AS(p, p);
#pragma unroll
    for (int r = 0; r < NDLY; ++r) {
      if (r > p) {
        const c32 fct = cdiv(AS(r, p), piv);
#pragma unroll
        for (int c = 0; c < NDLY; ++c) {
          if (c > p) AS(r, c) = csub(AS(r, c), cmul(fct, AS(p, c)));
        }
        bb[r] = csub(bb[r], cmul(fct, bb[p]));
      }
    }
  }
  c32 xs[NDLY];
#pragma unroll
  for (int r = NDLY - 1; r >= 0; --r) {
    c32 s = bb[r];
#pragma unroll
    for (int c = 0; c < NDLY; ++c) {
      if (c > r) s = csub(s, cmul(AS(r, c), xs[c]));
    }
    xs[r] = cdiv(s, AS(r, r));
  }

  c32 acc = { 0.0f, 0.0f };
#pragma unroll
  for (int d = 0; d < NDLY; ++d) {
    const c32 t = cmul(dvec[d], xs[d]);
    acc.re += t.re * out_g[d];
    acc.im += t.im * out_g[d];
  }
  transfer[f] = make_float2(acc.re, acc.im);
#undef AS
}

// ---------------------------------------------------------------------------
// K2: direct real iDFT (N=32000). One output sample per thread; spectrum
// chunked through LDS; exact phase via integer residue recurrence.
// Writes the zero-padded IR (hpad) and adds early_ir to the first 200 taps.
// ---------------------------------------------------------------------------
__global__ void fdn_idft_kernel(const float2* __restrict__ transfer,
                                const float* __restrict__ early_ir,
                                float* __restrict__ hpad) {
  __shared__ float2 ch[1024];
  const int n = blockIdx.x * 256 + threadIdx.x;
  float acc = 0.0f;
  for (int k0 = 0; k0 < NBINS; k0 += 1024) {
    const int cnt = min(1024, NBINS - k0);
    __syncthreads();
    for (int i = threadIdx.x; i < cnt; i += 256) ch[i] = transfer[k0 + i];
    __syncthreads();
    if (n < NFFT) {
      int m = (int)(((long long)k0 * (long long)n) % (long long)NFFT);
      for (int i = 0; i < cnt; ++i) {
        const int k = k0 + i;
        const float w = (k == 0 || k == NFFT / 2) ? 1.0f : 2.0f;
        float s, c;
        __sincosf((float)m * (TWO_PI_F / (float)NFFT), &s, &c);
        const float2 X = ch[i];
        acc += w * (X.x * c - X.y * s);
        m += n;
        if (m >= NFFT) m -= NFFT;
      }
    }
  }
  if (n < NFFT) {
    float v = acc * (1.0f / (float)NFFT);
    if (n < EARLY_LEN) v += early_ir[n];
    hpad[16 + n] = v;
  }
  if (n < 16) hpad[n] = 0.0f;                 // left zero pad
  if (n < 48) hpad[16 + NFFT + n] = 0.0f;     // right zero pad
}

// ---------------------------------------------------------------------------
// K3: zero-padded input image, 65536 floats per batch, signal at +16000.
// ---------------------------------------------------------------------------
__global__ void fdn_xpad_kernel(const float* __restrict__ audio,
                                float* __restrict__ xpad) {
  const int gid = blockIdx.x * 256 + threadIdx.x;     // 4 * 65536 threads
  const int b = gid >> 16;
  const int i = gid & (XPAD_STRIDE - 1);
  float v = 0.0f;
  if (i >= XPAD_LEFT && i < XPAD_LEFT + NFFT)
    v = audio[b * NFFT + (i - XPAD_LEFT)];
  xpad[gid] = v;
}

// ---------------------------------------------------------------------------
// K4: convolution as band-Toeplitz GEMM on V_WMMA_F32_16X16X4_F32.
//   D[i,j] += sum_k h[c-i+k] * x[u0+16j+k],  c = 4q, u0 = base + c - left
//   => out[base + 16j + i], taps swept exactly once over q = 0..8003.
// 8 waves per block, one 256-sample output tile per wave. The padded IR
// (125 KB, fits CDNA5's 320 KB LDS) is DMA'd into LDS by the Tensor Data
// Mover; per-iteration B operands come in as one global_load_b128 per lane.
// ---------------------------------------------------------------------------
__global__ void fdn_conv_wmma_kernel(const float* __restrict__ hpad,
                                     const float* __restrict__ xpad,
                                     float* __restrict__ out) {
  extern __shared__ float hsh[];                       // HPAD_LEN floats
  const int tid = threadIdx.x;

  // --- TDM: one descriptor DMAs the whole padded IR into LDS -------------
  // D# (ISA 8.3/8.4): 2D tensor, data_size=4B, dim0=tile0=HPAD_LEN, dim1=1.
  if (tid == 0) {
    const unsigned long long ga = (unsigned long long)(uintptr_t)hpad;
    const unsigned int lds_off = (unsigned int)(uintptr_t)&hsh[0];
    u32x4 g0;
    g0[0] = 1u;                                            // count=1 (valid)
    g0[1] = lds_off;                                       // lds_addr (bytes)
    g0[2] = (unsigned int)(ga & 0xFFFFFFFFull);            // global_addr[31:0]
    g0[3] = (unsigned int)((ga >> 32) & 0x01FFFFFFull)     // global_addr[56:32]
            | 0x80000000u;                                 // type=2 ("image")
    i32x8 g1;
    g1[0] = (int)(2u << 16);                               // data_size=4B
    g1[1] = (int)(((unsigned)HPAD_LEN & 0xFFFFu) << 16);   // tensor_dim0[15:0]
    g1[2] = (int)(((unsigned)HPAD_LEN >> 16) | (1u << 16));// dim0[31:16] | tensor_dim1=1
    g1[3] = (int)(((unsigned)HPAD_LEN & 0xFFFFu) << 16);   // dim1[31:16]=0 | tile_dim0
    g1[4] = 1;                                             // tile_dim1=1, tile_dim2=0
    g1[5] = HPAD_LEN;                                      // tensor_dim0_stride[31:0]
    g1[6] = (int)(((unsigned)HPAD_LEN & 0xFFFFu) << 16);   // stride0[47:32]=0 | stride1[15:0]
    g1[7] = 0;                                             // tensor_dim1_stride[47:16]=0
    const i32x4 gz4 = { 0, 0, 0, 0 };                      // groups 2/3 unused (<=2D)
    const i32x8 gz8 = { 0, 0, 0, 0, 0, 0, 0, 0 };          // trailing group (unused)
    __builtin_amdgcn_tensor_load_to_lds(g0, g1, gz4, gz4, gz8, 0);
    __builtin_amdgcn_s_wait_tensorcnt(0);
  }
  __syncthreads();

  const int wave = tid >> 5;
  const int lane = tid & 31;
  const int tile = blockIdx.x * 8 + wave;
  if (tile >= 500) return;                             // wave-uniform: EXEC stays all-ones

  const int b    = tile / 125;
  const int base = (tile % 125) * 256;
  const int lrow = lane & 15;                          // A row (M) / B col (N)
  const int hi   = lane >> 4;

  const float* __restrict__ xb = xpad + (size_t)b * XPAD_STRIDE;
  // A[m,k] = hpad[16 + 4q - m + k]; lane layout: VGPR0 holds K=0/2, VGPR1 K=1/3
  int ia = 16 - lrow + 2 * hi;
  // B[k,n] = xpad[base + 4q - left + 16n + k] (+XPAD_LEFT); rows {0,2}/{1,3} per lane half.
  // One b128 per lane spans offsets {0..3}; B.x = word0, B.y = word2.
  int g0i = base + (XPAD_LEFT - CONV_LEFT) + 16 * lrow + hi;

  v8f acc = { 0.0f, 0.0f, 0.0f, 0.0f, 0.0f, 0.0f, 0.0f, 0.0f };
#pragma unroll 4
  for (int q = 0; q < NQ; ++q) {
    const f32x4 xv = *(const f32x4*)(xb + g0i);        // 4B-aligned b128 (DWORD mode OK)
    v2f A, B;
    A.x = hsh[ia];
    A.y = hsh[ia + 1];
    B.x = xv.x;
    B.y = xv.z;
    acc = __builtin_amdgcn_wmma_f32_16x16x4_f32(
        /*neg_a=*/false, A, /*neg_b=*/false, B,
        /*c_mod=*/(short)0, acc, /*reuse_a=*/false, /*reuse_b=*/false);
    ia  += 4;
    g0i += 4;
  }

  // D lane layout: lane l, VGPR r -> (M = r + 8*hi, N = lrow) -> out[base + 16N + M]
  float* __restrict__ op = out + (size_t)b * NFFT + base + 16 * lrow + 8 * hi;
#pragma unroll
  for (int r = 0; r < 8; ++r) op[r] = acc[r];
}

// ---------------------------------------------------------------------------
extern "C" void kernel_launch(void* const* d_in, const int* in_sizes, int n_in,
                              void* d_out, int out_size, void* d_ws, size_t ws_size,
                              hipStream_t stream) {
  const float* audio = (const float*)d_in[0];   // [4,1,32000]
  const float* early = (const float*)d_in[1];   // [200]
  const float* in_g  = (const float*)d_in[2];   // [8]
  const float* out_g = (const float*)d_in[3];   // [8]
  const float* t0    = (const float*)d_in[4];   // [1]
  const float* alpha = (const float*)d_in[5];   // [1]
  const float* dval  = (const float*)d_in[6];   // [8]
  const float* dapl  = (const float*)d_in[7];   // [8,4]
  const float* gapl  = (const float*)d_in[8];   // [8,4]
  float* out = (float*)d_out;                   // [4,32000]

  float*  ws       = (float*)d_ws;
  float2* transfer = (float2*)ws;
  float*  hpad     = ws + WS_HPAD_OFF;
  float*  xpad     = ws + WS_XPAD_OFF;

  // K1: 16001 bins, 64 threads/block
  fdn_freq_kernel<<<(NBINS + 63) / 64, 64, 0, stream>>>(
      in_g, out_g, t0, alpha, dval, dapl, gapl, transfer);

  // K2: 32000 output samples
  fdn_idft_kernel<<<NFFT / 256, 256, 0, stream>>>(transfer, early, hpad);

  // K3: 4 * 65536 padded input elements
  fdn_xpad_kernel<<<(4 * XPAD_STRIDE) / 256, 256, 0, stream>>>(audio, xpad);

  // K4: 500 output tiles, 8 waves/block
  fdn_conv_wmma_kernel<<<63, 256, HPAD_LEN * sizeof(float), stream>>>(
      hpad, xpad, out);
}